// LookupFreeQuantizer_10428180595132
// MI455X (gfx1250) — compile-verified
//
#include <hip/hip_runtime.h>
#include <hip/hip_bf16.h>

typedef float v2f __attribute__((ext_vector_type(2)));
typedef float v4f __attribute__((ext_vector_type(4)));
typedef float v8f __attribute__((ext_vector_type(8)));

#define NPOS      65536      // 16*64*64 positions
#define NELEM     655360     // 16*10*64*64
#define OUT_LOSS  655360
#define OUT_PSE   655361
#define OUT_IDX   655362
#define ROW       68         // LDS row stride in floats (16B aligned, bank-spread)
#define WAVE_LDS  (32*ROW)   // floats per wave region

// ws layout: [0..1023] avg_probs accumulator, [1024] commit sum, [1025] per-sample-entropy sum

__global__ __launch_bounds__(128) void lfq_main(const float* __restrict__ z,
                                                float* __restrict__ out,
                                                float* __restrict__ ws) {
    __shared__ float prod[4 * WAVE_LDS];   // 4 waves * 32 pos * 68
    __shared__ float acc[1024];

    const int tid  = threadIdx.x;      // 0..127
    const int lane = tid & 31;
    const int wave = tid >> 5;         // 0..3
    const int m    = lane & 15;
    const int half = lane >> 4;

    for (int i = tid; i < 1024; i += 128) acc[i] = 0.0f;
    __syncthreads();

    float* myProd = &prod[wave * WAVE_LDS];
    const int waveGlobal = blockIdx.x * 4 + wave;   // 0..1023

    v8f c00 = {}, c01 = {}, c10 = {}, c11 = {};
    float commitAcc = 0.0f, pseAcc = 0.0f;

    #pragma unroll
    for (int it = 0; it < 2; ++it) {
        const int p  = waveGlobal * 64 + it * 32 + lane;   // position
        const int b  = p >> 12;
        const int hw = p & 4095;
        const float* zp = z   + b * 40960 + hw;
        float*       zq = out + b * 40960 + hw;

        // ---- per-position elementwise pass: sign, index, commitment, bit entropies, q_c ----
        float q[10];
        int idx = 0;
        #pragma unroll
        for (int c = 0; c < 10; ++c) {
            const float zv = zp[c * 4096];
            const bool  sp = zv > 0.0f;
            const float s  = sp ? 1.0f : -1.0f;
            zq[c * 4096] = s;
            idx |= ((int)sp) << c;
            const float d = s - zv;
            commitAcc += d * d;
            // per-bit prob of +1: q = sigmoid(400*z)  (exact factorization of the 1024-way softmax)
            const float ax   = fabsf(400.0f * zv);
            const float e    = __expf(-ax);          // e in (0,1]
            const float inv  = 1.0f / (1.0f + e);
            const float qmin = e * inv;              // minority probability
            q[c] = sp ? inv : qmin;
            // Bernoulli entropy (nats): H = qmin*|x| + log1p(e)
            pseAcc += qmin * ax + log1pf(e);
        }
        out[OUT_IDX + p] = (float)idx;

        // ---- partial products over bit halves: pLo (bits 0..4), pHi (bits 5..9) ----
        float pl[32], ph[32];
        pl[0] = 1.0f; ph[0] = 1.0f;
        #pragma unroll
        for (int c = 0; c < 5; ++c) {
            const int len = 1 << c;
            const float ql = q[c], qh = q[5 + c];
            #pragma unroll
            for (int j = len - 1; j >= 0; --j) {
                pl[j + len] = pl[j] * ql;          pl[j] *= (1.0f - ql);
                ph[j + len] = ph[j] * qh;          ph[j] *= (1.0f - qh);
            }
        }

        // ---- stage to LDS: row per position = [pHi[0..31] | pLo[0..31]] ----
        v4f* row = (v4f*)&myProd[lane * ROW];
        #pragma unroll
        for (int j = 0; j < 8; ++j)
            row[j] = (v4f){ph[4*j], ph[4*j+1], ph[4*j+2], ph[4*j+3]};
        #pragma unroll
        for (int j = 0; j < 8; ++j)
            row[8 + j] = (v4f){pl[4*j], pl[4*j+1], pl[4*j+2], pl[4*j+3]};

        __builtin_amdgcn_wave_barrier();   // same-wave DS ordering: stores before reads

        // ---- avg32x32 += pHi^T(32xK) * pLo(Kx32), K = 32 positions, via WMMA f32 16x16x4 ----
        // A layout (16x4 f32): lane m, VGPR0 = K even, VGPR1 = K odd; lane-half selects K pair.
        #pragma unroll
        for (int s = 0; s < 8; ++s) {
            const float* base = myProd + (4 * s + 2 * half) * ROW;
            v2f a0, a1, b0, b1;
            a0.x = base[m];            a0.y = base[ROW + m];
            a1.x = base[16 + m];       a1.y = base[ROW + 16 + m];
            b0.x = base[32 + m];       b0.y = base[ROW + 32 + m];
            b1.x = base[48 + m];       b1.y = base[ROW + 48 + m];
            c00 = __builtin_amdgcn_wmma_f32_16x16x4_f32(false, a0, false, b0, (short)0, c00, false, false);
            c01 = __builtin_amdgcn_wmma_f32_16x16x4_f32(false, a0, false, b1, (short)0, c01, false, false);
            c10 = __builtin_amdgcn_wmma_f32_16x16x4_f32(false, a1, false, b0, (short)0, c10, false, false);
            c11 = __builtin_amdgcn_wmma_f32_16x16x4_f32(false, a1, false, b1, (short)0, c11, false, false);
        }
    }

    // ---- dump accumulator tiles: C/D layout lane n = lane&15, M = r + 8*half ----
    #pragma unroll
    for (int r = 0; r < 8; ++r) {
        const int hi = r + 8 * half;
        atomicAdd(&acc[( hi       << 5) |  m      ], c00[r]);
        atomicAdd(&acc[( hi       << 5) | (16 + m)], c01[r]);
        atomicAdd(&acc[((16 + hi) << 5) |  m      ], c10[r]);
        atomicAdd(&acc[((16 + hi) << 5) | (16 + m)], c11[r]);
    }
    atomicAdd(&ws[1024], commitAcc);
    atomicAdd(&ws[1025], pseAcc);

    __syncthreads();
    for (int i = tid; i < 1024; i += 128) atomicAdd(&ws[i], acc[i]);
}

__global__ __launch_bounds__(256) void lfq_finalize(const float* __restrict__ ws,
                                                    float* __restrict__ out) {
    __shared__ float red[256];
    const int t = threadIdx.x;
    float s = 0.0f;
    for (int i = t; i < 1024; i += 256) {
        const float p = ws[i] * (1.0f / 65536.0f);   // avg_probs
        s += p * logf(p + 1e-5f);
    }
    red[t] = s;
    for (int off = 128; off > 0; off >>= 1) {
        __syncthreads();
        if (t < off) red[t] += red[t + off];
    }
    __syncthreads();
    if (t == 0) {
        const float avg_entropy = -red[0];                      // ENTROPY_GAMMA = 1.0
        const float commit      = 0.25f * ws[1024] / (float)NELEM;
        const float pse         = ws[1025] / (float)NPOS;
        const float loss        = commit + 0.1f * (pse - avg_entropy);
        out[OUT_LOSS] = loss;
        out[OUT_PSE]  = pse;
    }
}

extern "C" void kernel_launch(void* const* d_in, const int* in_sizes, int n_in,
                              void* d_out, int out_size, void* d_ws, size_t ws_size,
                              hipStream_t stream) {
    const float* z  = (const float*)d_in[0];
    float* out = (float*)d_out;
    float* ws  = (float*)d_ws;

    hipMemsetAsync(ws, 0, 1026 * sizeof(float), stream);
    lfq_main<<<256, 128, 0, stream>>>(z, out, ws);
    lfq_finalize<<<1, 256, 0, stream>>>(ws, out);
}